// EarthMoversDistanceLoss_72438918414881
// MI455X (gfx1250) — compile-verified
//
#include <hip/hip_runtime.h>

typedef __attribute__((ext_vector_type(2))) float v2f;
typedef __attribute__((ext_vector_type(4))) float v4f;
typedef __attribute__((ext_vector_type(8))) float v8f;

#define BINS 128
#define LDS_STRIDE 132      // 128 + 4 pad: removes 16-way bank conflicts on A-operand ds_load_b64
#define ROWS_PER_WAVE 16
#define WAVES_PER_BLOCK 4
#define THREADS_PER_BLOCK (WAVES_PER_BLOCK * 32)

// EMD loss: out[r] = sum_j cumsum(x[r]-y[r])[j]^2
// cumsum realized as triangular-matmul on V_WMMA_F32_16X16X4_F32 with a
// running carry matrix S (rowsums replicated across N via all-ones B).
__global__ __launch_bounds__(THREADS_PER_BLOCK)
void emd_wmma_f32_kernel(const float* __restrict__ x,
                         const float* __restrict__ y,
                         float* __restrict__ out,
                         int rows)
{
    __shared__ float lds[WAVES_PER_BLOCK * ROWS_PER_WAVE * LDS_STRIDE];

    const int lane    = threadIdx.x & 31;
    const int wave    = threadIdx.x >> 5;
    const int rowBase = (blockIdx.x * WAVES_PER_BLOCK + wave) * ROWS_PER_WAVE;
    if (rowBase >= rows) return;   // uniform per wave -> EXEC stays all-ones below

    float* tile = &lds[wave * ROWS_PER_WAVE * LDS_STRIDE];

    // ---- Stage diff = x - y into LDS (coalesced b128 loads, one row per iter) ----
    {
        const size_t gbase = (size_t)rowBase * BINS;
        #pragma unroll
        for (int it = 0; it < 16; ++it) {
            const int idx = (it * 32 + lane) * 4;     // 0..2044 step 4
            const int r   = idx >> 7;                 // idx / 128
            const int c   = idx & 127;
            v4f xv = *(const v4f*)(x + gbase + idx);
            v4f yv = *(const v4f*)(y + gbase + idx);
            v4f d  = xv - yv;
            *(v4f*)(tile + r * LDS_STRIDE + c) = d;
        }
    }

    // ---- Constant B operands (4x16 f32 layout: VGPR0 -> K=0|2, VGPR1 -> K=1|3) ----
    const int nloc = lane & 15;                 // N column within tile
    const int klo  = (lane < 16) ? 0 : 2;       // K rows held by this half-wave
    v2f btri[4];
    #pragma unroll
    for (int s = 0; s < 4; ++s) {
        // B[k][n] = (n >= k) for the diagonal (intra-block cumsum) tile
        btri[s].x = (nloc >= 4 * s + klo)     ? 1.0f : 0.0f;
        btri[s].y = (nloc >= 4 * s + klo + 1) ? 1.0f : 0.0f;
    }
    v2f bones; bones.x = 1.0f; bones.y = 1.0f;  // all-ones B: replicates row sums over N

    // A operand source: 16x4 f32 layout (lanes 0-15: M=lane,K=0,1; lanes 16-31: M=lane-16,K=2,3)
    const float* arow = tile + (lane & 15) * LDS_STRIDE + klo;

    v8f S   = {};   // carry: rowsums of K-blocks < t, replicated across N (C/D layout)
    v8f acc = {};   // per-lane sum of squared CDF values

    #pragma unroll
    for (int t = 0; t < 8; ++t) {
        v8f c = S;  // carry-in for this 16-column output tile
        #pragma unroll
        for (int s = 0; s < 4; ++s) {
            const int kk = 4 * t + s;
            v2f a = *(const v2f*)(arow + 4 * kk);   // ds_load_b64
            // intra-block triangular contribution
            c = __builtin_amdgcn_wmma_f32_16x16x4_f32(false, a, false, btri[s],
                                                      (short)0, c, false, false);
            // carry update: S += rowsum of this K-slab (replicated across N)
            S = __builtin_amdgcn_wmma_f32_16x16x4_f32(false, a, false, bones,
                                                      (short)0, S, false, false);
        }
        #pragma unroll
        for (int v = 0; v < 8; ++v) acc[v] += c[v] * c[v];
    }

    // ---- Reduce over N: rows m live in lanes 0-15, rows m+8 in lanes 16-31 ----
    float red[8];
    #pragma unroll
    for (int v = 0; v < 8; ++v) {
        float a = acc[v];
        a += __shfl_xor(a, 1);
        a += __shfl_xor(a, 2);
        a += __shfl_xor(a, 4);
        a += __shfl_xor(a, 8);
        red[v] = a;
    }
    if (lane == 0) {
        #pragma unroll
        for (int v = 0; v < 8; ++v) out[rowBase + v] = red[v];
    } else if (lane == 16) {
        #pragma unroll
        for (int v = 0; v < 8; ++v) out[rowBase + 8 + v] = red[v];
    }
}

extern "C" void kernel_launch(void* const* d_in, const int* in_sizes, int n_in,
                              void* d_out, int out_size, void* d_ws, size_t ws_size,
                              hipStream_t stream) {
    const float* x = (const float*)d_in[0];
    const float* y = (const float*)d_in[1];
    float* out = (float*)d_out;

    const int rows = in_sizes[0] / BINS;                       // 1048576
    const int rowsPerBlock = WAVES_PER_BLOCK * ROWS_PER_WAVE;  // 64
    const int blocks = (rows + rowsPerBlock - 1) / rowsPerBlock;

    hipLaunchKernelGGL(emd_wmma_f32_kernel, dim3(blocks), dim3(THREADS_PER_BLOCK),
                       0, stream, x, y, out, rows);
}